// BFM_18923625906658
// MI455X (gfx1250) — compile-verified
//
#include <hip/hip_runtime.h>
#include <math.h>

#define KD 128

typedef __attribute__((ext_vector_type(2))) float v2f;
typedef __attribute__((ext_vector_type(8))) float v8f;

// Workspace layout (floats):
//   [0]            bias accumulator
//   [1]            sq_sum accumulator
//   [8 .. 8+3*KD)  V matrix, row-major 3 x 128: row0=u_vec, row1=t_vec, row2=s
#define WS_BIAS  0
#define WS_SQ    1
#define WS_V     8
#define WS_TOTAL (WS_V + 3 * KD)

__global__ __launch_bounds__(256) void fm_zero(float* ws) {
    int i = blockIdx.x * 256 + threadIdx.x;
    if (i < WS_TOTAL) ws[i] = 0.0f;
}

__device__ __forceinline__ void fm_accum(int i, float v, int n, int m,
                                         const float* __restrict__ w_bias,
                                         const float* __restrict__ uV,
                                         const float* __restrict__ tV,
                                         const float* __restrict__ bV,
                                         float* ws) {
    // bias contribution (only at nonzeros -> never read w_bias densely)
    atomicAdd(&ws[WS_BIAS], v * w_bias[i]);
    if (i < n) {
        const float* r = uV + (size_t)i * KD;
        float* acc = ws + WS_V;
#pragma unroll 8
        for (int k = 0; k < KD; ++k) atomicAdd(&acc[k], v * r[k]);
    } else if (i < n + m) {
        const float* r = tV + (size_t)(i - n) * KD;
        float* acc = ws + WS_V + KD;
#pragma unroll 8
        for (int k = 0; k < KD; ++k) atomicAdd(&acc[k], v * r[k]);
    } else {
        const float* r = bV + (size_t)(i - n - m) * KD;
        float* acc = ws + WS_V + 2 * KD;
        float sq = 0.0f;
#pragma unroll 8
        for (int k = 0; k < KD; ++k) {
            float e = r[k];
            atomicAdd(&acc[k], v * e);
            sq = fmaf(e, e, sq);
        }
        atomicAdd(&ws[WS_SQ], v * sq);
    }
}

// One pass over x[:p], float4-vectorized (global_load_b128). Nonzeros are
// rare (~52), so the gather/atomic path is cold.
__global__ __launch_bounds__(256) void fm_scan(const float* __restrict__ x,
                                               const float* __restrict__ w_bias,
                                               const float* __restrict__ uV,
                                               const float* __restrict__ tV,
                                               const float* __restrict__ bV,
                                               float* ws, int n, int m, int p) {
    int t = blockIdx.x * 256 + threadIdx.x;
    int base = t * 4;
    if (base + 3 < p) {
        const float4 q = *(const float4*)(x + base);
        if (q.x != 0.0f) fm_accum(base + 0, q.x, n, m, w_bias, uV, tV, bV, ws);
        if (q.y != 0.0f) fm_accum(base + 1, q.y, n, m, w_bias, uV, tV, bV, ws);
        if (q.z != 0.0f) fm_accum(base + 2, q.z, n, m, w_bias, uV, tV, bV, ws);
        if (q.w != 0.0f) fm_accum(base + 3, q.w, n, m, w_bias, uV, tV, bV, ws);
    } else {
        for (int i = base; i < p; ++i) {
            float v = x[i];
            if (v != 0.0f) fm_accum(i, v, n, m, w_bias, uV, tV, bV, ws);
        }
    }
}

// Single-wave finalizer: Gram matrix of V (3x128, rows u,t,s) via
// V_WMMA_F32_16X16X4_F32, K looped in steps of 4. A (16x4) and B (4x16)
// operands have identical per-lane contents for a symmetric product:
//   lane L, vgpr v : V[L%16][k0 + v + (L<16 ? 0 : 2)]
// C layout: vgpr r, lanes 0-15 -> C[r][lane]; lanes 16-31 -> C[r+8][lane-16].
__global__ __launch_bounds__(32) void fm_final(const float* __restrict__ ws,
                                               const float* __restrict__ w0,
                                               const float* __restrict__ delta,
                                               float* __restrict__ out) {
    const int lane = threadIdx.x;          // wave32
    const int row  = lane & 15;            // A row (M) / B col (N)
    const int koff = (lane < 16) ? 0 : 2;  // K sub-offset for this half-wave
    const float* V = ws + WS_V;            // 3 x 128 row-major; rows 3..15 = 0

    v8f c = {0.f, 0.f, 0.f, 0.f, 0.f, 0.f, 0.f, 0.f};
#pragma unroll
    for (int k0 = 0; k0 < KD; k0 += 4) {
        v2f a;
        if (row < 3) {
            a.x = V[row * KD + k0 + koff + 0];
            a.y = V[row * KD + k0 + koff + 1];
        } else {
            a.x = 0.0f;
            a.y = 0.0f;
        }
        // D = A*B + C ; B operand == A operand (symmetric Gram product)
        c = __builtin_amdgcn_wmma_f32_16x16x4_f32(
                /*neg_a=*/false, a, /*neg_b=*/false, a,
                /*c_mod=*/(short)0, c, /*reuse_a=*/false, /*reuse_b=*/false);
    }

    // Route the 4 needed Gram entries to lane 0 via LDS.
    __shared__ float cm[3][16];
    if (lane < 16) {
        cm[0][lane] = c[0];   // C[0][lane]
        cm[1][lane] = c[1];   // C[1][lane]
        cm[2][lane] = c[2];   // C[2][lane]
    }
    __syncthreads();

    if (lane == 0) {
        float ut = cm[0][1];  // u . t
        float ts = cm[1][2];  // t . s
        float ss = cm[2][2];  // s . s
        float us = cm[0][2];  // u . s
        float bias = ws[WS_BIAS];
        float sq   = ws[WS_SQ];
        float y = w0[0] + bias + ut + ts + 0.5f * (ss - sq) + us;
        float z = y * delta[0];
        // -log_sigmoid(z) = softplus(-z), numerically stable
        float a = -z;
        out[0] = fmaxf(a, 0.0f) + log1pf(expf(-fabsf(a)));
    }
}

extern "C" void kernel_launch(void* const* d_in, const int* in_sizes, int n_in,
                              void* d_out, int out_size, void* d_ws, size_t ws_size,
                              hipStream_t stream) {
    // Input order: x, delta, pmi, w_0, w_bias, u_V, t_V, b_V
    const float* x      = (const float*)d_in[0];
    const float* delta  = (const float*)d_in[1];
    // d_in[2] = pmi (unused by reference forward)
    const float* w0     = (const float*)d_in[3];
    const float* w_bias = (const float*)d_in[4];
    const float* uV     = (const float*)d_in[5];
    const float* tV     = (const float*)d_in[6];
    const float* bV     = (const float*)d_in[7];

    const int n = in_sizes[5] / KD;   // u_V rows
    const int m = in_sizes[6] / KD;   // t_V rows
    const int p = n + 2 * m;

    float* ws = (float*)d_ws;
    float* out = (float*)d_out;

    fm_zero<<<(WS_TOTAL + 255) / 256, 256, 0, stream>>>(ws);

    const int nquad = (p + 3) / 4;
    fm_scan<<<(nquad + 255) / 256, 256, 0, stream>>>(x, w_bias, uV, tV, bV, ws,
                                                     n, m, p);

    fm_final<<<1, 32, 0, stream>>>(ws, w0, delta, out);
}